// TopKDiceLoss_3212635537498
// MI455X (gfx1250) — compile-verified
//
#include <hip/hip_runtime.h>

#define TDL_SENTINEL 0xFFFFFFFFu
#define TDL_B 32

__device__ __forceinline__ unsigned lds_lo32(const void* p) {
  // low 32 bits of a flat shared-aperture address == LDS byte address
  return (unsigned)(size_t)p;
}

// ---------------- zero small workspace ----------------
__global__ void tdl_zero(unsigned* __restrict__ p, int n) {
  for (int i = blockIdx.x * blockDim.x + threadIdx.x; i < n;
       i += gridDim.x * blockDim.x)
    p[i] = 0u;
}

// ---------------- histogram pass (radix select digit) ----------------
// FIRST: also counts n_fg and (optionally) writes key/prob cache.
// FROM_KEYS: refine pass streaming cached keys only.
template <bool FIRST, bool FROM_KEYS>
__global__ __launch_bounds__(256) void tdl_hist_pass(
    const float* __restrict__ logits, const int* __restrict__ target,
    const float* __restrict__ eps, unsigned* __restrict__ keys,
    float* __restrict__ probs, unsigned* __restrict__ hist,
    unsigned* __restrict__ nfg, const unsigned* __restrict__ prefix,
    int shift, int N, int chunk) {
  const int b = blockIdx.y;
  const int tid = threadIdx.x;
  const int start = blockIdx.x * chunk;

  __shared__ unsigned lh[256];
  __shared__ unsigned cntS;
  lh[tid] = 0u;
  if (tid == 0) cntS = 0u;
  __syncthreads();

  const unsigned pref = FIRST ? 0u : prefix[b];
  unsigned cnt = 0u;

  if (FROM_KEYS) {
    const unsigned* kb = keys + (size_t)b * N;
    for (int t0 = start; t0 < start + chunk; t0 += 1024) {
      const int idx = t0 + tid * 4;
      uint4 k4 = *(const uint4*)(kb + idx);
      unsigned ks[4] = {k4.x, k4.y, k4.z, k4.w};
#pragma unroll
      for (int j = 0; j < 4; ++j) {
        unsigned key = ks[j];
        if (key != TDL_SENTINEL && ((key >> (shift + 8)) == pref))
          atomicAdd(&lh[(key >> shift) & 255u], 1u);
      }
    }
  } else {
    __shared__ float st0[1024];
    __shared__ float st1[1024];
    const float* lbase = logits + (size_t)b * 2u * (size_t)N;
    const unsigned long long sbase = (unsigned long long)(size_t)lbase;
    const unsigned lds0 = lds_lo32(st0) + (unsigned)tid * 16u;
    const unsigned lds1 = lds_lo32(st1) + (unsigned)tid * 16u;
    for (int t0 = start; t0 < start + chunk; t0 += 1024) {
      const int idx = t0 + tid * 4;
      const unsigned off0 = (unsigned)idx * 4u;
      const unsigned off1 = off0 + (unsigned)N * 4u;
      // CDNA5 async global->LDS copies (ASYNCcnt-tracked), 16B per lane
      asm volatile("global_load_async_to_lds_b128 %0, %1, %2"
                   :: "v"(lds0), "v"(off0), "s"(sbase) : "memory");
      asm volatile("global_load_async_to_lds_b128 %0, %1, %2"
                   :: "v"(lds1), "v"(off1), "s"(sbase) : "memory");
      int4 tg = *(const int4*)(target + (size_t)b * N + idx);
      float4 ep = *(const float4*)(eps + (size_t)b * N + idx);
      asm volatile("s_wait_asynccnt 0" ::: "memory");
      __syncthreads();
      int tgv[4] = {tg.x, tg.y, tg.z, tg.w};
      float epv[4] = {ep.x, ep.y, ep.z, ep.w};
      unsigned kk[4];
      float pp[4];
#pragma unroll
      for (int j = 0; j < 4; ++j) {
        float l0 = st0[tid * 4 + j];
        float l1 = st1[tid * 4 + j];
        float p = 1.0f / (1.0f + __expf(l0 - l1));  // softmax class-1
        bool fg = (tgv[j] == 1);
        float tp = p * ((float)tgv[j] + epv[j] * 1e-6f);
        unsigned key = __float_as_uint(tp);  // tp>=0: bits are monotone
        pp[j] = p;
        kk[j] = fg ? key : TDL_SENTINEL;
        if (fg) {
          cnt++;
          if (FIRST || ((key >> (shift + 8)) == pref))
            atomicAdd(&lh[(key >> shift) & 255u], 1u);
        }
      }
      if (FIRST && keys) {
        *(uint4*)(keys + (size_t)b * N + idx) =
            make_uint4(kk[0], kk[1], kk[2], kk[3]);
        *(float4*)(probs + (size_t)b * N + idx) =
            make_float4(pp[0], pp[1], pp[2], pp[3]);
      }
      __syncthreads();
    }
  }

  if (FIRST && cnt) atomicAdd(&cntS, cnt);
  __syncthreads();
  unsigned v = lh[tid];
  if (v) atomicAdd(&hist[b * 256 + tid], v);  // u32 adds: order-independent
  if (FIRST && tid == 0 && cntS) atomicAdd(&nfg[b], cntS);
}

// ---------------- per-batch digit selection + hist re-zero ----------------
__global__ __launch_bounds__(256) void tdl_scan(
    unsigned* __restrict__ hist, const unsigned* __restrict__ nfg,
    unsigned* __restrict__ krem, unsigned* __restrict__ prefix,
    unsigned* __restrict__ thrbits, int pass) {
  const int b = blockIdx.x;
  unsigned* h = hist + b * 256;
  if (threadIdx.x == 0) {
    unsigned k;
    if (pass == 0) {
      unsigned n = nfg[b];
      k = (unsigned)((float)n * 0.5f);  // int(n_fg * 50/100), trunc
      if (k < 1u) k = 1u;
    } else {
      k = krem[b];
    }
    unsigned cum = 0u, sel = 255u;
    for (unsigned bin = 0; bin < 256u; ++bin) {
      unsigned c = h[bin];
      if (k <= cum + c) { sel = bin; break; }
      cum += c;
    }
    k -= cum;
    krem[b] = k;
    unsigned pref = (pass == 0) ? sel : ((prefix[b] << 8) | sel);
    prefix[b] = pref;
    if (pass == 3)
      thrbits[b] = (nfg[b] == 0u) ? 0x7F800000u /* +inf: ignore nothing */
                                  : pref;
  }
  __syncthreads();
  h[threadIdx.x] = 0u;  // ready for next pass / next launch
}

// ---------------- final masked dice sums (deterministic partials) ---------
template <bool FROM_KEYS>
__global__ __launch_bounds__(256) void tdl_final_pass(
    const float* __restrict__ logits, const int* __restrict__ target,
    const float* __restrict__ eps, const unsigned* __restrict__ keys,
    const float* __restrict__ probs, const unsigned* __restrict__ thrbits,
    float* __restrict__ interP, float* __restrict__ uniP, int N, int chunk) {
  const int b = blockIdx.y;
  const int tid = threadIdx.x;
  const int start = blockIdx.x * chunk;
  const unsigned tb = thrbits[b];
  float isum = 0.f, usum = 0.f;

  if (FROM_KEYS) {
    const unsigned* kb = keys + (size_t)b * N;
    const float* pb = probs + (size_t)b * N;
    for (int t0 = start; t0 < start + chunk; t0 += 1024) {
      const int idx = t0 + tid * 4;
      uint4 k4 = *(const uint4*)(kb + idx);
      float4 p4 = *(const float4*)(pb + idx);
      unsigned ks[4] = {k4.x, k4.y, k4.z, k4.w};
      float ps[4] = {p4.x, p4.y, p4.z, p4.w};
#pragma unroll
      for (int j = 0; j < 4; ++j) {
        bool fg = (ks[j] != TDL_SENTINEL);
        bool ignore = fg && (ks[j] > tb);  // tp > thr via bit compare
        if (!ignore) {
          usum += ps[j];                            // sum(probs * mask)
          if (fg) { usum += 1.f; isum += ps[j]; }   // target part + inters.
        }
      }
    }
  } else {
    __shared__ float st0[1024];
    __shared__ float st1[1024];
    const float* lbase = logits + (size_t)b * 2u * (size_t)N;
    const unsigned long long sbase = (unsigned long long)(size_t)lbase;
    const unsigned lds0 = lds_lo32(st0) + (unsigned)tid * 16u;
    const unsigned lds1 = lds_lo32(st1) + (unsigned)tid * 16u;
    for (int t0 = start; t0 < start + chunk; t0 += 1024) {
      const int idx = t0 + tid * 4;
      const unsigned off0 = (unsigned)idx * 4u;
      const unsigned off1 = off0 + (unsigned)N * 4u;
      asm volatile("global_load_async_to_lds_b128 %0, %1, %2"
                   :: "v"(lds0), "v"(off0), "s"(sbase) : "memory");
      asm volatile("global_load_async_to_lds_b128 %0, %1, %2"
                   :: "v"(lds1), "v"(off1), "s"(sbase) : "memory");
      int4 tg = *(const int4*)(target + (size_t)b * N + idx);
      float4 ep = *(const float4*)(eps + (size_t)b * N + idx);
      asm volatile("s_wait_asynccnt 0" ::: "memory");
      __syncthreads();
      int tgv[4] = {tg.x, tg.y, tg.z, tg.w};
      float epv[4] = {ep.x, ep.y, ep.z, ep.w};
#pragma unroll
      for (int j = 0; j < 4; ++j) {
        float l0 = st0[tid * 4 + j];
        float l1 = st1[tid * 4 + j];
        float p = 1.0f / (1.0f + __expf(l0 - l1));
        bool fg = (tgv[j] == 1);
        float tp = p * ((float)tgv[j] + epv[j] * 1e-6f);
        unsigned key = __float_as_uint(tp);
        bool ignore = fg && (key > tb);
        if (!ignore) {
          usum += p;
          if (fg) { usum += 1.f; isum += p; }
        }
      }
      __syncthreads();
    }
  }

  // deterministic in-block tree reduction; one partial slot per block
  __shared__ float rs[256];
  rs[tid] = isum;
  __syncthreads();
  for (int s = 128; s > 0; s >>= 1) {
    if (tid < s) rs[tid] += rs[tid + s];
    __syncthreads();
  }
  if (tid == 0) interP[(size_t)b * gridDim.x + blockIdx.x] = rs[0];
  __syncthreads();
  rs[tid] = usum;
  __syncthreads();
  for (int s = 128; s > 0; s >>= 1) {
    if (tid < s) rs[tid] += rs[tid + s];
    __syncthreads();
  }
  if (tid == 0) uniP[(size_t)b * gridDim.x + blockIdx.x] = rs[0];
}

// ---------------- dice + mean (one wave32, fixed-order fold) -------------
__global__ void tdl_finish(const float* __restrict__ interP,
                           const float* __restrict__ uniP, int bpb,
                           float* __restrict__ out) {
  const int b = threadIdx.x;  // 32 lanes == 32 batches
  float I = 0.f, U = 0.f;
  for (int i = 0; i < bpb; ++i) {  // fixed serial order -> deterministic
    I += interP[(size_t)b * bpb + i];
    U += uniP[(size_t)b * bpb + i];
  }
  float dice = (U == 0.f) ? 1.f : (2.f * I) / fmaxf(U, 1e-6f);
  for (int o = 16; o > 0; o >>= 1) dice += __shfl_down(dice, o, 32);
  if (b == 0) out[0] = 1.f - dice * (1.f / (float)TDL_B);
}

extern "C" void kernel_launch(void* const* d_in, const int* in_sizes, int n_in,
                              void* d_out, int out_size, void* d_ws,
                              size_t ws_size, hipStream_t stream) {
  const float* logits = (const float*)d_in[0];
  const int* target = (const int*)d_in[1];
  const float* eps = (const float*)d_in[2];
  const int B = TDL_B;
  const int N = in_sizes[1] / B;  // 262144

  const int chunk = 4096;     // elements per block (4 x 1024-elem tiles)
  const int bpb = N / chunk;  // 64 blocks per batch
  dim3 grid(bpb, B);

  // ws: hist[B*256] | nfg[B] | krem[B] | prefix[B] | thrbits[B]
  //     | interP[B*bpb] | uniP[B*bpb] | (optional) keys[B*N] | probs[B*N]
  unsigned* hist = (unsigned*)d_ws;
  unsigned* nfg = hist + B * 256;
  unsigned* krem = nfg + B;
  unsigned* prefix = krem + B;
  unsigned* thrbits = prefix + B;
  float* interP = (float*)(thrbits + B);
  float* uniP = interP + (size_t)B * bpb;
  const int zeroWords = B * 256 + 4 * B;  // hist + counters only
  size_t smallBytes = (size_t)(B * 256 + 4 * B) * 4 + (size_t)B * bpb * 8;

  size_t cacheOff = (smallBytes + 255) & ~(size_t)255;
  size_t cacheBytes = (size_t)B * (size_t)N * 8;
  bool cache = (ws_size >= cacheOff + cacheBytes);
  unsigned* keys = cache ? (unsigned*)((char*)d_ws + cacheOff) : nullptr;
  float* probs = cache ? (float*)(keys + (size_t)B * N) : nullptr;

  tdl_zero<<<32, 256, 0, stream>>>(hist, zeroWords);

  // pass 0: digit = bits[31:24], also n_fg (+ optional key/prob cache)
  tdl_hist_pass<true, false><<<grid, 256, 0, stream>>>(
      logits, target, eps, keys, probs, hist, nfg, prefix, 24, N, chunk);
  tdl_scan<<<B, 256, 0, stream>>>(hist, nfg, krem, prefix, thrbits, 0);

  for (int pass = 1; pass < 4; ++pass) {
    int shift = 24 - 8 * pass;
    if (cache)
      tdl_hist_pass<false, true><<<grid, 256, 0, stream>>>(
          logits, target, eps, keys, probs, hist, nfg, prefix, shift, N, chunk);
    else
      tdl_hist_pass<false, false><<<grid, 256, 0, stream>>>(
          logits, target, eps, nullptr, nullptr, hist, nfg, prefix, shift, N,
          chunk);
    tdl_scan<<<B, 256, 0, stream>>>(hist, nfg, krem, prefix, thrbits, pass);
  }

  if (cache)
    tdl_final_pass<true><<<grid, 256, 0, stream>>>(
        logits, target, eps, keys, probs, thrbits, interP, uniP, N, chunk);
  else
    tdl_final_pass<false><<<grid, 256, 0, stream>>>(
        logits, target, eps, keys, probs, thrbits, interP, uniP, N, chunk);

  tdl_finish<<<1, 32, 0, stream>>>(interP, uniP, bpb, (float*)d_out);
}